// GATGraphRegressor_52269751992502
// MI455X (gfx1250) — compile-verified
//
#include <hip/hip_runtime.h>
#include <hip/hip_bf16.h>
#include <math.h>

typedef __attribute__((ext_vector_type(2))) float v2f;
typedef __attribute__((ext_vector_type(8))) float v8f;

#define HDIM 64
#define FIN  128

// exact branchless ELU: for x>0, exp(min(x,0))-1 == 0 and max picks x;
// for x<=0, exp(x)-1 >= x (convexity), so max picks exp(x)-1.
__device__ __forceinline__ float eluf(float x)
{
    return fmaxf(x, expf(fminf(x, 0.0f)) - 1.0f);
}
__device__ __forceinline__ float lrelu(float x) { return x > 0.0f ? x : 0.2f * x; }

// out[N][64] = (elu?)(A[N][K]) @ W[K][64] + bias ; one wave per 16-row strip.
// WMMA f32 16x16x4, fragment layouts per CDNA5 ISA 7.12.2:
//   A: lanes 0-15 -> M=lane, v0=K0,v1=K1 ; lanes 16-31 -> v0=K2,v1=K3
//   B: lanes 0-15 -> N=lane, v0=K0,v1=K1 ; lanes 16-31 -> v0=K2,v1=K3
//   C/D: VGPR i -> row i (lanes 0-15) / row i+8 (lanes 16-31), col = lane&15
template <bool ELU>
__global__ void __launch_bounds__(128)
gat_gemm_wmma(const float* __restrict__ A, const float* __restrict__ W,
              const float* __restrict__ bias, float* __restrict__ out,
              int nrows, int K)
{
    const int lane  = threadIdx.x & 31;
    const int wave  = threadIdx.x >> 5;
    const int strip = blockIdx.x * (blockDim.x >> 5) + wave;
    if (strip * 16 >= nrows) return;                 // wave-uniform guard
    const int m0   = strip * 16;
    const int mrow = m0 + (lane & 15);
    const int koff = (lane >> 4) << 1;               // 0 or 2
    const int nlo  = lane & 15;
    const float* arow = A + (size_t)mrow * K + koff;

    v8f acc0 = {}, acc1 = {}, acc2 = {}, acc3 = {};
    for (int k = 0; k < K; k += 4) {
        float2 av = *(const float2*)(arow + k);
        if (ELU) { av.x = eluf(av.x); av.y = eluf(av.y); }
        v2f a; a.x = av.x; a.y = av.y;

        const float* w0 = W + (size_t)(k + koff) * HDIM;  // row k+koff
        const float* w1 = w0 + HDIM;                      // row k+koff+1
        v2f b0; b0.x = w0[nlo];      b0.y = w1[nlo];
        v2f b1; b1.x = w0[nlo + 16]; b1.y = w1[nlo + 16];
        v2f b2; b2.x = w0[nlo + 32]; b2.y = w1[nlo + 32];
        v2f b3; b3.x = w0[nlo + 48]; b3.y = w1[nlo + 48];

        acc0 = __builtin_amdgcn_wmma_f32_16x16x4_f32(false, a, false, b0, (short)0, acc0, false, false);
        acc1 = __builtin_amdgcn_wmma_f32_16x16x4_f32(false, a, false, b1, (short)0, acc1, false, false);
        acc2 = __builtin_amdgcn_wmma_f32_16x16x4_f32(false, a, false, b2, (short)0, acc2, false, false);
        acc3 = __builtin_amdgcn_wmma_f32_16x16x4_f32(false, a, false, b3, (short)0, acc3, false, false);
    }

    const int rbase = m0 + ((lane >> 4) << 3);
    const float bi0 = bias[nlo], bi1 = bias[nlo + 16], bi2 = bias[nlo + 32], bi3 = bias[nlo + 48];
    #pragma unroll
    for (int i = 0; i < 8; ++i) {
        float* orow = out + (size_t)(rbase + i) * HDIM + nlo;
        orow[0]  = acc0[i] + bi0;
        orow[16] = acc1[i] + bi1;
        orow[32] = acc2[i] + bi2;
        orow[48] = acc3[i] + bi3;
    }
}

// ai[n] = h[n,:] . att[0:64] ; aj[n] = h[n,:] . att[64:128] ; one wave per node
__global__ void __launch_bounds__(256)
att_dots(const float* __restrict__ h, const float* __restrict__ att,
         float* __restrict__ ai, float* __restrict__ aj, int n)
{
    const int lane = threadIdx.x & 31;
    const int node = blockIdx.x * (blockDim.x >> 5) + (threadIdx.x >> 5);
    if (node >= n) return;
    const float* hr = h + (size_t)node * HDIM;
    float h0 = hr[lane], h1 = hr[lane + 32];
    float pi = h0 * att[lane]      + h1 * att[lane + 32];
    float pj = h0 * att[lane + 64] + h1 * att[lane + 96];
    #pragma unroll
    for (int off = 16; off > 0; off >>= 1) {
        pi += __shfl_xor(pi, off, 32);
        pj += __shfl_xor(pj, off, 32);
    }
    if (lane == 0) { ai[node] = pi; aj[node] = pj; }
}

__global__ void __launch_bounds__(256)
init_nodes(float* __restrict__ agg, unsigned int* __restrict__ m,
           float* __restrict__ denom, int n)
{
    int idx = blockIdx.x * blockDim.x + threadIdx.x;
    if (idx < n * HDIM) agg[idx] = 0.0f;
    if (idx < n) { m[idx] = 0xFF800000u; denom[idx] = 0.0f; }  // m = -inf
}

__global__ void __launch_bounds__(256)
edge_max(const int* __restrict__ src, const int* __restrict__ dst,
         const float* __restrict__ ai, const float* __restrict__ aj,
         float* __restrict__ m, int E, int n)
{
    int e = blockIdx.x * blockDim.x + threadIdx.x;
    if (e >= E + n) return;
    int s, d;
    if (e < E) { s = src[e]; d = dst[e]; } else { s = d = e - E; }
    float sc = lrelu(ai[d] + aj[s]);
    int bits = __float_as_int(sc);
    if (sc >= 0.0f) atomicMax((int*)m + d, bits);
    else            atomicMin((unsigned int*)m + d, (unsigned int)bits);
}

__global__ void __launch_bounds__(256)
edge_sum(const int* __restrict__ src, const int* __restrict__ dst,
         const float* __restrict__ ai, const float* __restrict__ aj,
         const float* __restrict__ m, float* __restrict__ denom, int E, int n)
{
    int e = blockIdx.x * blockDim.x + threadIdx.x;
    if (e >= E + n) return;
    int s, d;
    if (e < E) { s = src[e]; d = dst[e]; } else { s = d = e - E; }
    float sc = lrelu(ai[d] + aj[s]);
    atomicAdd(denom + d, expf(sc - m[d]));
}

// one wave per edge: lane reads float2 of h[src], scatters 2 atomic adds to agg[dst]
__global__ void __launch_bounds__(256)
edge_agg(const int* __restrict__ src, const int* __restrict__ dst,
         const float* __restrict__ ai, const float* __restrict__ aj,
         const float* __restrict__ m, const float* __restrict__ denom,
         const float* __restrict__ h, float* __restrict__ agg, int E, int n)
{
    const int lane = threadIdx.x & 31;
    const int e = blockIdx.x * (blockDim.x >> 5) + (threadIdx.x >> 5);
    if (e >= E + n) return;
    int s, d;
    if (e < E) { s = src[e]; d = dst[e]; } else { s = d = e - E; }
    float sc = lrelu(ai[d] + aj[s]);
    float alpha = expf(sc - m[d]) / denom[d];
    float2 hv = *(const float2*)(h + (size_t)s * HDIM + lane * 2);
    float* ad = agg + (size_t)d * HDIM + lane * 2;
    atomicAdd(ad,     alpha * hv.x);
    atomicAdd(ad + 1, alpha * hv.y);
}

__global__ void __launch_bounds__(256)
zero_kernel(float* __restrict__ p, int n)
{
    int i = blockIdx.x * blockDim.x + threadIdx.x;
    if (i < n) p[i] = 0.0f;
}

// 64 threads per node: elu + segment-sum into pooled[G][64], count per graph
__global__ void __launch_bounds__(256)
pool_kernel(const float* __restrict__ agg, const int* __restrict__ batch,
            float* __restrict__ pooled, float* __restrict__ cnt, int n)
{
    const int node = blockIdx.x * (blockDim.x >> 6) + (threadIdx.x >> 6);
    const int col  = threadIdx.x & 63;
    if (node >= n) return;
    int g = batch[node];
    atomicAdd(pooled + (size_t)g * HDIM + col, eluf(agg[(size_t)node * HDIM + col]));
    if (col == 0) atomicAdd(cnt + g, 1.0f);
}

__global__ void __launch_bounds__(256)
final_kernel(const float* __restrict__ pooled, const float* __restrict__ cnt,
             const float* __restrict__ Wfc, const float* __restrict__ bfc,
             float* __restrict__ out, int gcount)
{
    int g = blockIdx.x * blockDim.x + threadIdx.x;
    if (g >= gcount) return;
    float s = 0.0f;
    #pragma unroll
    for (int c = 0; c < HDIM; ++c) s += pooled[(size_t)g * HDIM + c] * Wfc[c];
    out[g] = s / cnt[g] + bfc[0];
}

extern "C" void kernel_launch(void* const* d_in, const int* in_sizes, int n_in,
                              void* d_out, int out_size, void* d_ws, size_t ws_size,
                              hipStream_t stream)
{
    const float* x    = (const float*)d_in[0];
    const int*   ei   = (const int*)d_in[1];
    const int*   batch= (const int*)d_in[2];
    const float* W1   = (const float*)d_in[3];
    const float* b1   = (const float*)d_in[4];
    const float* att1 = (const float*)d_in[5];
    const float* W2   = (const float*)d_in[6];
    const float* b2   = (const float*)d_in[7];
    const float* att2 = (const float*)d_in[8];
    const float* Wfc  = (const float*)d_in[9];
    const float* bfc  = (const float*)d_in[10];

    const int N = in_sizes[0] / FIN;
    const int E = in_sizes[1] / 2;
    const int G = out_size;               // OUT == 1
    const int* src = ei;                  // edge_index[0]
    const int* dst = ei + E;              // edge_index[1]

    // workspace layout (floats)
    float* bufA   = (float*)d_ws;                 // N*64  (h_lin)
    float* bufB   = bufA + (size_t)N * HDIM;      // N*64  (agg)
    float* ai     = bufB + (size_t)N * HDIM;      // N
    float* aj     = ai + N;                       // N
    float* mbuf   = aj + N;                       // N
    float* denom  = mbuf + N;                     // N
    float* pooled = denom + N;                    // G*64
    float* cnt    = pooled + (size_t)G * HDIM;    // G

    const int ET = E + N;
    const int strips = (N + 15) / 16;
    dim3 gemmGrid((strips + 3) / 4), gemmBlk(128);
    const int nodeWaveBlocks = (N + 7) / 8;
    const int initBlocks = (N * HDIM + 255) / 256;
    const int edgeBlocks = (ET + 255) / 256;
    const int aggBlocks  = (ET + 7) / 8;
    const int poolBlocks = (N + 3) / 4;

    // ---- layer 1 ----
    gat_gemm_wmma<false><<<gemmGrid, gemmBlk, 0, stream>>>(x, W1, b1, bufA, N, FIN);
    att_dots<<<nodeWaveBlocks, 256, 0, stream>>>(bufA, att1, ai, aj, N);
    init_nodes<<<initBlocks, 256, 0, stream>>>(bufB, (unsigned int*)mbuf, denom, N);
    edge_max<<<edgeBlocks, 256, 0, stream>>>(src, dst, ai, aj, mbuf, E, N);
    edge_sum<<<edgeBlocks, 256, 0, stream>>>(src, dst, ai, aj, mbuf, denom, E, N);
    edge_agg<<<aggBlocks, 256, 0, stream>>>(src, dst, ai, aj, mbuf, denom, bufA, bufB, E, N);

    // ---- layer 2 (ELU fused into GEMM A-load) ----
    gat_gemm_wmma<true><<<gemmGrid, gemmBlk, 0, stream>>>(bufB, W2, b2, bufA, N, HDIM);
    att_dots<<<nodeWaveBlocks, 256, 0, stream>>>(bufA, att2, ai, aj, N);
    init_nodes<<<initBlocks, 256, 0, stream>>>(bufB, (unsigned int*)mbuf, denom, N);
    edge_max<<<edgeBlocks, 256, 0, stream>>>(src, dst, ai, aj, mbuf, E, N);
    edge_sum<<<edgeBlocks, 256, 0, stream>>>(src, dst, ai, aj, mbuf, denom, E, N);
    edge_agg<<<aggBlocks, 256, 0, stream>>>(src, dst, ai, aj, mbuf, denom, bufA, bufB, E, N);

    // ---- pool + head (ELU fused into pooling) ----
    zero_kernel<<<(G * HDIM + G + 255) / 256, 256, 0, stream>>>(pooled, G * HDIM + G);
    pool_kernel<<<poolBlocks, 256, 0, stream>>>(bufB, batch, pooled, cnt, N);
    final_kernel<<<1, 256, 0, stream>>>(pooled, cnt, Wfc, bfc, (float*)d_out, G);
}